// TPForth_18511309046145
// MI455X (gfx1250) — compile-verified
//
#include <hip/hip_runtime.h>

typedef __attribute__((ext_vector_type(2))) float v2f;
typedef __attribute__((ext_vector_type(8))) float v8f;

#define X1DIM 156
#define WDIM  7128
#define OUTDIM 156

// weight block offsets (floats) within one sample's 7128-float weight slice
#define OW00  0      // 48x48
#define OW01  2304   // 48x10
#define OW10  2784   // 10x10
#define OW110 2884   // 10x48
#define OW112 3364   // 10x10
#define OW12  3464   // 10x10
#define OW20  3564   // 10x10
#define OW211 3664   // 10x10
#define OW213 3764   // 10x48
#define OW22  4244   // 10x10
#define OW30  4344   // 48x48
#define OW31  6648   // 48x10

// norms
#define NORM_0E 0.13130643f   // sqrt(1/58)
#define NORM_1O 0.19611614f   // sqrt(3/78)
#define NORM_1E 0.19611614f   // sqrt(3/78)
#define NORM_0O 0.13130643f   // sqrt(1/58)
#define RS3     0.57735027f   // 1/sqrt(3)
#define ISQ6    0.40824829f   // 1/sqrt(6)  (w3j(1,1,1) ~ eps_ijk/sqrt(6))
#define CA      0.31622777f   // 1/sqrt(10)
#define CB      0.18257419f   // 1/sqrt(30)
#define CC      0.36514837f   // 2/sqrt(30)

static __device__ __forceinline__ v8f wmma4(v2f a, v2f b, v8f c) {
  // D = A(16x4) * B(4x16) + C, fp32 throughout
  return __builtin_amdgcn_wmma_f32_16x16x4_f32(false, a, false, b, (short)0, c,
                                               false, false);
}

// GEMV t[w] = sum_u sp[u] * Wg[u*WN + w]; acc[nc] += scale * t[nc*16 + lane]
// for lanes 0..15. sp must be zero-padded to a multiple of 4 entries and
// 8-byte aligned. All operand loads are full-exec; masking is value-level.
template <int U, int WN>
static __device__ __forceinline__ void gemv_wmma(const float* __restrict__ Wg,
                                                 const float* sp, int lane,
                                                 float scale, float* acc) {
  const int m  = lane & 15;
  const int hi = lane >> 4;
  constexpr int KC = (U + 3) / 4;
  constexpr int NC = (WN + 15) / 16;
  const bool row0 = (m == 0);

  // ---- A operands once per path: ds_load_b64 + cndmask, full exec ----
  const float* spl = sp + hi * 2;  // lanes16-31 hold K=2,3 of each chunk
  v2f av[KC];
#pragma unroll
  for (int kc = 0; kc < KC; ++kc) {
    v2f sv = *(const v2f*)(spl + kc * 4);
    av[kc].x = row0 ? sv.x : 0.0f;
    av[kc].y = row0 ? sv.y : 0.0f;
  }

  // per-lane weight base: clamped column keeps all loads in-bounds;
  // garbage in D columns >= WN is never read (columns are independent).
  const int colc = (WN >= 16) ? m : ((m < WN) ? m : (WN - 1));
  const float* wl = Wg + colc + hi * 2 * WN;  // all further offsets are immediates

#pragma unroll
  for (int nc = 0; nc < NC; ++nc) {
    // batch all B loads for this N-chunk before the WMMA chain
    v2f bv[KC];
#pragma unroll
    for (int kc = 0; kc < KC; ++kc) {
      bv[kc].x = __builtin_nontemporal_load(wl + kc * 4 * WN + nc * 16);
      bv[kc].y = __builtin_nontemporal_load(wl + kc * 4 * WN + WN + nc * 16);
    }
    // two independent accumulation chains to halve WMMA dependency depth
    v8f c0 = {0.f, 0.f, 0.f, 0.f, 0.f, 0.f, 0.f, 0.f};
    v8f c1 = {0.f, 0.f, 0.f, 0.f, 0.f, 0.f, 0.f, 0.f};
#pragma unroll
    for (int kc = 0; kc < KC; ++kc) {
      if (kc & 1)
        c1 = wmma4(av[kc], bv[kc], c1);
      else
        c0 = wmma4(av[kc], bv[kc], c0);
    }
    if (lane < 16) acc[nc] += scale * (c0[0] + c1[0]);
  }
}

// 3-column GEMM: T[w][k] = sum_u S[u*SST + k] * Wg[u*10 + w], U=10, WN=10.
// acc3[k] += scale * T[lane][k] for lanes 0..9.
template <int SST>
static __device__ __forceinline__ void gemm3_wmma(const float* __restrict__ Wg,
                                                  const float* sp, int lane,
                                                  float scale, float* acc3) {
  const int m  = lane & 15;
  const int hi = lane >> 4;
  const bool am = (m < 3);
  const int mc = am ? m : 0;          // clamped in-range source column
  const float* spl = sp + hi * 2 * SST + mc;
  const int colc = (m < 10) ? m : 9;  // clamped in-range weight column
  const float* wl = Wg + colc + hi * 2 * 10;

  v2f av[3], bv[3];
#pragma unroll
  for (int kc = 0; kc < 3; ++kc) {
    const float s0 = spl[kc * 4 * SST];
    const float s1 = spl[kc * 4 * SST + SST];
    const bool ok0 = am && (kc * 4 + hi * 2 < 10);
    const bool ok1 = am && (kc * 4 + hi * 2 + 1 < 10);
    av[kc].x = ok0 ? s0 : 0.0f;
    av[kc].y = ok1 ? s1 : 0.0f;
    // rows 10,11 read into the following weight block (in-bounds); their
    // contribution is zeroed through A.
    bv[kc].x = __builtin_nontemporal_load(wl + kc * 40);
    bv[kc].y = __builtin_nontemporal_load(wl + kc * 40 + 10);
  }
  v8f c = {0.f, 0.f, 0.f, 0.f, 0.f, 0.f, 0.f, 0.f};
#pragma unroll
  for (int kc = 0; kc < 3; ++kc) c = wmma4(av[kc], bv[kc], c);
  if (lane < 10) {  // rows M=0..2 of D live in c[0..2], lanes 0..15
    acc3[0] += scale * c[0];
    acc3[1] += scale * c[1];
    acc3[2] += scale * c[2];
  }
}

__global__ __launch_bounds__(256) void tp_forth_kernel(
    const float* __restrict__ x1, const float* __restrict__ x2,
    const float* __restrict__ wts, float* __restrict__ out, int B) {
  __shared__ float lds[8 * 384];  // per-wave: x1[156] | x2[9] pad | pre[216]
  const int lane = threadIdx.x & 31;
  const int wv = threadIdx.x >> 5;
  float* sh  = lds + wv * 384;
  float* x1s = sh;        // 156
  float* x2s = sh + 156;  // 9
  float* pre = sh + 168;  // s110[12] | s213[12] | S112[48] | S12[48] | S211[48] | S22[48]

  const long z0 = (long)blockIdx.x * 8 + wv;
  const bool active = (z0 < B);
  const long z = active ? z0 : 0;
  const float* x1p = x1 + z * X1DIM;
  const float* x2p = x2 + z * 9;
  const float* wp  = wts + (size_t)z * WDIM;

  for (int t = lane; t < X1DIM; t += 32) x1s[t] = x1p[t];
  if (lane < 9) x2s[lane] = x2p[lane];
  __syncthreads();

  // ---- precompute per-u source vectors (lanes 0..11; 10,11 write zero pad) ----
  if (lane < 12) {
    float s110v = 0.f, s213v = 0.f;
    float c112[3] = {0.f, 0.f, 0.f}, c211[3] = {0.f, 0.f, 0.f};
    float c12[3]  = {0.f, 0.f, 0.f}, c22[3]  = {0.f, 0.f, 0.f};
    if (lane < 10) {
      const float p0 = x2s[1], p1 = x2s[2], p2 = x2s[3];
      const float a0 = x1s[48 + lane * 3 + 0];
      const float a1 = x1s[48 + lane * 3 + 1];
      const float a2 = x1s[48 + lane * 3 + 2];
      const float e0 = x1s[78 + lane * 3 + 0];
      const float e1 = x1s[78 + lane * 3 + 1];
      const float e2 = x1s[78 + lane * 3 + 2];
      s110v = a0 * p0 + a1 * p1 + a2 * p2;
      s213v = e0 * p0 + e1 * p1 + e2 * p2;
      // w3j(1,1,1): cross / sqrt(6)
      c112[0] = (a1 * p2 - a2 * p1) * ISQ6;
      c112[1] = (a2 * p0 - a0 * p2) * ISQ6;
      c112[2] = (a0 * p1 - a1 * p0) * ISQ6;
      c211[0] = (e1 * p2 - e2 * p1) * ISQ6;
      c211[1] = (e2 * p0 - e0 * p2) * ISQ6;
      c211[2] = (e0 * p1 - e1 * p0) * ISQ6;
      // w3j(1,2,1): symmetric traceless matrix N built from x2_2e (basis y,z,x)
      const float q0 = x2s[4], q1 = x2s[5], q2 = x2s[6], q3 = x2s[7], q4 = x2s[8];
      const float N00 = -CB * q2 - CA * q4;
      const float N11 = CC * q2;
      const float N22 = -CB * q2 + CA * q4;
      const float N01 = CA * q1, N02 = CA * q0, N12 = CA * q3;
      c12[0] = a0 * N00 + a1 * N01 + a2 * N02;
      c12[1] = a0 * N01 + a1 * N11 + a2 * N12;
      c12[2] = a0 * N02 + a1 * N12 + a2 * N22;
      c22[0] = e0 * N00 + e1 * N01 + e2 * N02;
      c22[1] = e0 * N01 + e1 * N11 + e2 * N12;
      c22[2] = e0 * N02 + e1 * N12 + e2 * N22;
    }
    pre[0 + lane]  = s110v;
    pre[12 + lane] = s213v;
#pragma unroll
    for (int k = 0; k < 3; ++k) {
      pre[24 + lane * 4 + k]  = c112[k];
      pre[72 + lane * 4 + k]  = c12[k];
      pre[120 + lane * 4 + k] = c211[k];
      pre[168 + lane * 4 + k] = c22[k];
    }
  }
  __syncthreads();

  const float s0e = x2s[0];
  const float p0 = x2s[1], p1 = x2s[2], p2 = x2s[3];

  // register-resident output accumulators (each output owned by one lane)
  float r0e_a[3] = {0.f, 0.f, 0.f};
  float r0o_a[3] = {0.f, 0.f, 0.f};
  float r1o_a[3] = {0.f, 0.f, 0.f};
  float r1e_a[3] = {0.f, 0.f, 0.f};

  // r0e = N0e*x2_0e * (x1_0e @ w00) + N0e*RS3 * (s110 @ w110)
  gemv_wmma<48, 48>(wp + OW00, x1s, lane, NORM_0E * s0e, r0e_a);
  gemv_wmma<12, 48>(wp + OW110, pre + 0, lane, NORM_0E * RS3, r0e_a);
  // r0o = N0o*RS3 * (s213 @ w213) + N0o*x2_0e * (x1_0o @ w30)
  gemv_wmma<12, 48>(wp + OW213, pre + 12, lane, NORM_0O * RS3, r0o_a);
  gemv_wmma<48, 48>(wp + OW30, x1s + 108, lane, NORM_0O * s0e, r0o_a);

  // r1o[w][j] += N1o*RS3 * (x1_0e @ w01)[w] * x2_1o[j]
  {
    float t[1] = {0.f};
    gemv_wmma<48, 10>(wp + OW01, x1s, lane, NORM_1O * RS3, t);
    r1o_a[0] += t[0] * p0;
    r1o_a[1] += t[0] * p1;
    r1o_a[2] += t[0] * p2;
  }
  // r1e[w][j] += N1e*RS3 * (x1_0o @ w31)[w] * x2_1o[j]
  {
    float t[1] = {0.f};
    gemv_wmma<48, 10>(wp + OW31, x1s + 108, lane, NORM_1E * RS3, t);
    r1e_a[0] += t[0] * p0;
    r1e_a[1] += t[0] * p1;
    r1e_a[2] += t[0] * p2;
  }

  // 10x10 paths with 3 shared source columns
  gemm3_wmma<3>(wp + OW10, x1s + 48, lane, NORM_1O * RS3 * s0e, r1o_a);   // w10
  gemm3_wmma<4>(wp + OW112, pre + 24, lane, NORM_1E, r1e_a);              // w112
  gemm3_wmma<4>(wp + OW12, pre + 72, lane, NORM_1O, r1o_a);               // w12
  gemm3_wmma<3>(wp + OW20, x1s + 78, lane, NORM_1E * RS3 * s0e, r1e_a);   // w20
  gemm3_wmma<4>(wp + OW211, pre + 120, lane, NORM_1O, r1o_a);             // w211
  gemm3_wmma<4>(wp + OW22, pre + 168, lane, NORM_1E, r1e_a);              // w22

  // ---- store: out = [r0e(48) | r1o(30) | r1e(30) | r0o(48)] ----
  if (active) {
    float* op = out + (size_t)z * OUTDIM;
    if (lane < 16) {
      op[lane]       = r0e_a[0];
      op[16 + lane]  = r0e_a[1];
      op[32 + lane]  = r0e_a[2];
      op[108 + lane] = r0o_a[0];
      op[124 + lane] = r0o_a[1];
      op[140 + lane] = r0o_a[2];
    }
    if (lane < 10) {
      op[48 + lane * 3 + 0] = r1o_a[0];
      op[48 + lane * 3 + 1] = r1o_a[1];
      op[48 + lane * 3 + 2] = r1o_a[2];
      op[78 + lane * 3 + 0] = r1e_a[0];
      op[78 + lane * 3 + 1] = r1e_a[1];
      op[78 + lane * 3 + 2] = r1e_a[2];
    }
  }
}

extern "C" void kernel_launch(void* const* d_in, const int* in_sizes, int n_in,
                              void* d_out, int out_size, void* d_ws,
                              size_t ws_size, hipStream_t stream) {
  (void)n_in; (void)d_ws; (void)ws_size; (void)out_size;
  const float* x1 = (const float*)d_in[0];
  const float* x2 = (const float*)d_in[1];
  const float* w  = (const float*)d_in[2];
  float* out = (float*)d_out;
  const int B = in_sizes[0] / X1DIM;
  const int grid = (B + 7) / 8;
  hipLaunchKernelGGL(tp_forth_kernel, dim3(grid), dim3(256), 0, stream,
                     x1, x2, w, out, B);
}